// Seq2Seq_40673340293428
// MI455X (gfx1250) — compile-verified
//
#include <hip/hip_runtime.h>
#include <hip/hip_bf16.h>
#include <math.h>

#define H 128
#define G3 384          // 3*H
#define T 2048
#define V 50257
#define STEPS 10
#define LOGITS_BLOCK 256
#define LOGITS_GRID ((V + LOGITS_BLOCK - 1) / LOGITS_BLOCK)   // 197

typedef float v2f __attribute__((ext_vector_type(2)));
typedef float v8f __attribute__((ext_vector_type(8)));

__device__ __forceinline__ float bflo(unsigned int u) {   // bf16 in u[15:0]
    return __uint_as_float(u << 16);
}
__device__ __forceinline__ float bfhi(unsigned int u) {   // bf16 in u[31:16]
    return __uint_as_float(u & 0xffff0000u);
}
__device__ __forceinline__ float sigm(float x) {
    return 1.0f / (1.0f + __expf(-x));
}

// ---------------------------------------------------------------------------
// 1) Gather token embeddings: xemb[t][:] = enc_emb[x[t]][:]
// ---------------------------------------------------------------------------
__global__ void embed_kernel(const int* __restrict__ x,
                             const float* __restrict__ emb,
                             float* __restrict__ xemb) {
    int t = blockIdx.x;
    int tok = x[t];                      // x is (T,1) int32
    const float* src = emb + (size_t)tok * H;
    for (int i = threadIdx.x; i < H; i += blockDim.x)
        xemb[(size_t)t * H + i] = src[i];
}

// ---------------------------------------------------------------------------
// 2) Gi0[T][G3] = Xemb[T][H] * Wih0^T[H][G3] + bih0   via V_WMMA_F32_16X16X4_F32
//    Each wave computes one 16x16 output tile; K-loop of 32 WMMAs.
//    A (16x4 f32): lanes 0-15 -> K=k+0,k+1 ; lanes 16-31 -> K=k+2,k+3
//    B (4x16 f32): VGPR0/1 mirror A's K split; N = lane&15; B[kk][n] = W[n][kk]
//    D (16x16 f32): c[r] -> M = r + 8*(lane>=16), N = lane&15
// ---------------------------------------------------------------------------
__global__ void gi0_wmma_kernel(const float* __restrict__ Xe,   // [T][H]
                                const float* __restrict__ W,    // [G3][H]
                                const float* __restrict__ bias, // [G3]
                                float* __restrict__ Gi) {       // [T][G3]
    const int lane = threadIdx.x & 31;
    const int wave = threadIdx.x >> 5;
    const int m0 = blockIdx.x * 16;
    const int n0 = (blockIdx.y * 4 + wave) * 16;
    const int half = lane >> 4;      // 0 or 1
    const int l15  = lane & 15;

    const float* arow = Xe + (size_t)(m0 + l15) * H + half * 2;
    const float* brow = W  + (size_t)(n0 + l15) * H + half * 2;

    v8f c = {};
    #pragma unroll 4
    for (int k = 0; k < H; k += 4) {
        v2f a = { arow[k], arow[k + 1] };
        v2f b = { brow[k], brow[k + 1] };
        c = __builtin_amdgcn_wmma_f32_16x16x4_f32(
                false, a, false, b, (short)0, c, false, false);
    }
    #pragma unroll
    for (int r = 0; r < 8; ++r) {
        int m = m0 + r + half * 8;
        int n = n0 + l15;
        Gi[(size_t)m * G3 + n] = c[r] + bias[n];
    }
}

// ---------------------------------------------------------------------------
// 3) Encoder recurrence: single workgroup (384 threads = 12 wave32 on 1 WGP),
//    recurrent weights held in LDS as bf16 (3 * 96KB = 288KB < 320KB WGP LDS).
//    Critical-path serial loop: ds_load_b128 weight reads (8 bf16/load),
//    ds_load_b128 h reads, ping-pong h buffers -> 4 barriers per timestep.
// ---------------------------------------------------------------------------
#define ENC_F32_WORDS (4 * H + 2 * G3 + 3 * G3)
#define ENC_SMEM_BYTES (3 * G3 * H * 2 + ENC_F32_WORDS * 4)

__global__ void enc_rnn_kernel(const float* __restrict__ Gi0,   // [T][G3] (bih0 folded in)
                               const float* __restrict__ Whh0,
                               const float* __restrict__ bhh0,
                               const float* __restrict__ Wih1,
                               const float* __restrict__ bih1,
                               const float* __restrict__ Whh1,
                               const float* __restrict__ bhh1,
                               float* __restrict__ hstate) {    // [2*H] out
    extern __shared__ char smem[];
    unsigned short* w0  = (unsigned short*)smem;     // Whh0 bf16 [G3*H]
    unsigned short* wi1 = w0  + G3 * H;              // Wih1 bf16
    unsigned short* w1  = wi1 + G3 * H;              // Whh1 bf16
    float* fb    = (float*)(w1 + G3 * H);            // 16B-aligned (294912 % 16 == 0)
    float* h1buf = fb;             // [2*H] ping-pong
    float* h2buf = h1buf + 2 * H;  // [2*H] ping-pong
    float* gh    = h2buf + 2 * H;  // [G3]
    float* gi    = gh + G3;        // [G3]
    float* b0    = gi + G3;        // [G3] bhh0
    float* bi1   = b0 + G3;        // [G3] bih1
    float* b1    = bi1 + G3;       // [G3] bhh1

    const int tid = threadIdx.x;   // blockDim.x == 384

    for (int i = tid; i < G3 * H; i += blockDim.x) {
        w0[i]  = (unsigned short)(__float_as_uint(Whh0[i]) >> 16);
        wi1[i] = (unsigned short)(__float_as_uint(Wih1[i]) >> 16);
        w1[i]  = (unsigned short)(__float_as_uint(Whh1[i]) >> 16);
    }
    for (int i = tid; i < G3; i += blockDim.x) {
        b0[i] = bhh0[i]; bi1[i] = bih1[i]; b1[i] = bhh1[i];
    }
    for (int i = tid; i < 2 * H; i += blockDim.x) { h1buf[i] = 0.0f; h2buf[i] = 0.0f; }
    __syncthreads();

    const uint4* wr0  = (const uint4*)(w0  + tid * H);   // 16 x uint4 per row
    const uint4* wri1 = (const uint4*)(wi1 + tid * H);
    const uint4* wrh1 = (const uint4*)(w1  + tid * H);

    for (int t = 0; t < T; ++t) {
        const int cur = t & 1;
        float* h1c = h1buf + cur * H;
        float* h1n = h1buf + (cur ^ 1) * H;
        float* h2c = h2buf + cur * H;
        float* h2n = h2buf + (cur ^ 1) * H;

        // ---- layer 0 matvec: gh = Whh0 @ h1 + bhh0 (one row per thread)
        {
            float acc = b0[tid];
            const float4* hv = (const float4*)h1c;
            #pragma unroll
            for (int k = 0; k < H / 8; ++k) {
                uint4 w = wr0[k];
                float4 ha = hv[2 * k], hb = hv[2 * k + 1];
                acc += bflo(w.x) * ha.x + bfhi(w.x) * ha.y
                     + bflo(w.y) * ha.z + bfhi(w.y) * ha.w
                     + bflo(w.z) * hb.x + bfhi(w.z) * hb.y
                     + bflo(w.w) * hb.z + bfhi(w.w) * hb.w;
            }
            gh[tid] = acc;
        }
        __syncthreads();
        // ---- layer 0 gates -> h1 (next buffer)
        if (tid < H) {
            const float* g = Gi0 + (size_t)t * G3;
            float r = sigm(g[tid]          + gh[tid]);
            float z = sigm(g[H + tid]      + gh[H + tid]);
            float n = tanhf(g[2 * H + tid] + r * gh[2 * H + tid]);
            h1n[tid] = (1.0f - z) * n + z * h1c[tid];
        }
        __syncthreads();
        // ---- layer 1 matvecs: gi = Wih1 @ h1new + bih1 ; gh = Whh1 @ h2 + bhh1
        {
            float ai = bi1[tid], ah = b1[tid];
            const float4* hv1 = (const float4*)h1n;
            const float4* hv2 = (const float4*)h2c;
            #pragma unroll
            for (int k = 0; k < H / 8; ++k) {
                uint4 wa = wri1[k];
                uint4 wb = wrh1[k];
                float4 pa = hv1[2 * k], pb = hv1[2 * k + 1];
                float4 qa = hv2[2 * k], qb = hv2[2 * k + 1];
                ai += bflo(wa.x) * pa.x + bfhi(wa.x) * pa.y
                    + bflo(wa.y) * pa.z + bfhi(wa.y) * pa.w
                    + bflo(wa.z) * pb.x + bfhi(wa.z) * pb.y
                    + bflo(wa.w) * pb.z + bfhi(wa.w) * pb.w;
                ah += bflo(wb.x) * qa.x + bfhi(wb.x) * qa.y
                    + bflo(wb.y) * qa.z + bfhi(wb.y) * qa.w
                    + bflo(wb.z) * qb.x + bfhi(wb.z) * qb.y
                    + bflo(wb.w) * qb.z + bfhi(wb.w) * qb.w;
            }
            gi[tid] = ai; gh[tid] = ah;
        }
        __syncthreads();
        // ---- layer 1 gates -> h2 (next buffer)
        if (tid < H) {
            float r = sigm(gi[tid]          + gh[tid]);
            float z = sigm(gi[H + tid]      + gh[H + tid]);
            float n = tanhf(gi[2 * H + tid] + r * gh[2 * H + tid]);
            h2n[tid] = (1.0f - z) * n + z * h2c[tid];
        }
        __syncthreads();
    }
    const int fin = T & 1;    // T even -> final state in buffer 0
    if (tid < H) {
        hstate[tid]     = h1buf[fin * H + tid];
        hstate[H + tid] = h2buf[fin * H + tid];
    }
}

// ---------------------------------------------------------------------------
// 4) Decoder helpers
// ---------------------------------------------------------------------------
__global__ void init_yid_kernel(const int* __restrict__ sid, int* __restrict__ yid) {
    if (threadIdx.x == 0) yid[0] = sid[0];
}

__global__ void dec_rnn_kernel(const float* __restrict__ demb,
                               const int*   __restrict__ yid,
                               const float* __restrict__ Wih0, const float* __restrict__ Whh0,
                               const float* __restrict__ bih0, const float* __restrict__ bhh0,
                               const float* __restrict__ Wih1, const float* __restrict__ Whh1,
                               const float* __restrict__ bih1, const float* __restrict__ bhh1,
                               float* __restrict__ dstate) {    // [2*H] in/out
    __shared__ __align__(16) float e[H];
    __shared__ __align__(16) float d1[H];
    __shared__ __align__(16) float d2[H];
    __shared__ float gi[G3], gh[G3];
    const int tid = threadIdx.x;     // blockDim.x == 384
    if (tid < H) {
        e[tid]  = demb[(size_t)yid[0] * H + tid];
        d1[tid] = dstate[tid];
        d2[tid] = dstate[H + tid];
    }
    __syncthreads();
    // layer 0
    {
        float ai = bih0[tid], ah = bhh0[tid];
        const float* wi = Wih0 + (size_t)tid * H;
        const float* wh = Whh0 + (size_t)tid * H;
        for (int k = 0; k < H; k += 4) {
            float4 a = *(const float4*)(wi + k);
            float4 b = *(const float4*)(wh + k);
            ai += a.x * e[k]  + a.y * e[k + 1]  + a.z * e[k + 2]  + a.w * e[k + 3];
            ah += b.x * d1[k] + b.y * d1[k + 1] + b.z * d1[k + 2] + b.w * d1[k + 3];
        }
        gi[tid] = ai; gh[tid] = ah;
    }
    __syncthreads();
    if (tid < H) {
        float r = sigm(gi[tid]          + gh[tid]);
        float z = sigm(gi[H + tid]      + gh[H + tid]);
        float n = tanhf(gi[2 * H + tid] + r * gh[2 * H + tid]);
        e[tid] = (1.0f - z) * n + z * d1[tid];          // new d1 (e no longer needed)
    }
    __syncthreads();
    if (tid < H) d1[tid] = e[tid];
    __syncthreads();
    // layer 1
    {
        float ai = bih1[tid], ah = bhh1[tid];
        const float* wi = Wih1 + (size_t)tid * H;
        const float* wh = Whh1 + (size_t)tid * H;
        for (int k = 0; k < H; k += 4) {
            float4 a = *(const float4*)(wi + k);
            float4 b = *(const float4*)(wh + k);
            ai += a.x * d1[k] + a.y * d1[k + 1] + a.z * d1[k + 2] + a.w * d1[k + 3];
            ah += b.x * d2[k] + b.y * d2[k + 1] + b.z * d2[k + 2] + b.w * d2[k + 3];
        }
        gi[tid] = ai; gh[tid] = ah;
    }
    __syncthreads();
    if (tid < H) {
        float r = sigm(gi[tid]          + gh[tid]);
        float z = sigm(gi[H + tid]      + gh[H + tid]);
        float n = tanhf(gi[2 * H + tid] + r * gh[2 * H + tid]);
        e[tid] = (1.0f - z) * n + z * d2[tid];          // new d2
    }
    __syncthreads();
    if (tid < H) { dstate[tid] = d1[tid]; dstate[H + tid] = e[tid]; }
}

// 5) logits = out_W @ d2 + out_b, per-block argmax
__global__ void logits_kernel(const float* __restrict__ outW,
                              const float* __restrict__ outb,
                              const float* __restrict__ dstate,
                              float* __restrict__ bmax, int* __restrict__ bidx) {
    __shared__ __align__(16) float sh[H];
    __shared__ float smax[LOGITS_BLOCK];
    __shared__ int   sidx[LOGITS_BLOCK];
    const int tid = threadIdx.x;
    const int row = blockIdx.x * LOGITS_BLOCK + tid;
    if (tid < H) sh[tid] = dstate[H + tid];             // d2
    __syncthreads();
    float val = -INFINITY;
    if (row < V) {
        float acc = outb[row];
        const float* wr = outW + (size_t)row * H;
        __builtin_prefetch(wr + H, 0, 1);               // global_prefetch_b8
        for (int k = 0; k < H; k += 4) {
            float4 w = *(const float4*)(wr + k);
            acc += w.x * sh[k] + w.y * sh[k + 1] + w.z * sh[k + 2] + w.w * sh[k + 3];
        }
        val = acc;
    }
    smax[tid] = val; sidx[tid] = row;
    __syncthreads();
    for (int s = LOGITS_BLOCK / 2; s > 0; s >>= 1) {
        if (tid < s) {
            bool take = (smax[tid + s] > smax[tid]) ||
                        (smax[tid + s] == smax[tid] && sidx[tid + s] < sidx[tid]);
            if (take) { smax[tid] = smax[tid + s]; sidx[tid] = sidx[tid + s]; }
        }
        __syncthreads();
    }
    if (tid == 0) { bmax[blockIdx.x] = smax[0]; bidx[blockIdx.x] = sidx[0]; }
}

// 6) final argmax over block partials; emit token (first-max tie-break)
__global__ void argmax_reduce_kernel(const float* __restrict__ bmax,
                                     const int* __restrict__ bidx,
                                     int nblocks, int* __restrict__ yid,
                                     float* __restrict__ out, int step) {
    __shared__ float smax[256];
    __shared__ int   sidx[256];
    const int tid = threadIdx.x;
    float best = -INFINITY; int bi = 0x7fffffff;
    for (int i = tid; i < nblocks; i += 256) {
        float v = bmax[i]; int ix = bidx[i];
        if (v > best || (v == best && ix < bi)) { best = v; bi = ix; }
    }
    smax[tid] = best; sidx[tid] = bi;
    __syncthreads();
    for (int s = 128; s > 0; s >>= 1) {
        if (tid < s) {
            bool take = (smax[tid + s] > smax[tid]) ||
                        (smax[tid + s] == smax[tid] && sidx[tid + s] < sidx[tid]);
            if (take) { smax[tid] = smax[tid + s]; sidx[tid] = sidx[tid + s]; }
        }
        __syncthreads();
    }
    if (tid == 0) { yid[0] = sidx[0]; out[step] = (float)sidx[0]; }
}

// ---------------------------------------------------------------------------
extern "C" void kernel_launch(void* const* d_in, const int* in_sizes, int n_in,
                              void* d_out, int out_size, void* d_ws, size_t ws_size,
                              hipStream_t stream) {
    const int*   x        = (const int*)  d_in[0];
    const int*   sid      = (const int*)  d_in[1];
    /* eid (d_in[2]) is unused: reference scans a fixed STEPS with no early stop */
    const float* enc_emb  = (const float*)d_in[3];
    const float* dec_emb  = (const float*)d_in[4];
    const float* enc_Wih0 = (const float*)d_in[5];
    const float* enc_Whh0 = (const float*)d_in[6];
    const float* enc_bih0 = (const float*)d_in[7];
    const float* enc_bhh0 = (const float*)d_in[8];
    const float* enc_Wih1 = (const float*)d_in[9];
    const float* enc_Whh1 = (const float*)d_in[10];
    const float* enc_bih1 = (const float*)d_in[11];
    const float* enc_bhh1 = (const float*)d_in[12];
    const float* dec_Wih0 = (const float*)d_in[13];
    const float* dec_Whh0 = (const float*)d_in[14];
    const float* dec_bih0 = (const float*)d_in[15];
    const float* dec_bhh0 = (const float*)d_in[16];
    const float* dec_Wih1 = (const float*)d_in[17];
    const float* dec_Whh1 = (const float*)d_in[18];
    const float* dec_bih1 = (const float*)d_in[19];
    const float* dec_bhh1 = (const float*)d_in[20];
    const float* out_W    = (const float*)d_in[21];
    const float* out_b    = (const float*)d_in[22];
    float* out = (float*)d_out;

    // workspace carve-up
    float* xemb   = (float*)d_ws;                    // T*H
    float* gi0    = xemb + (size_t)T * H;            // T*G3
    float* hstate = gi0 + (size_t)T * G3;            // 2*H (enc h -> dec d)
    float* bmax   = hstate + 2 * H;                  // LOGITS_GRID (<256)
    int*   bidx   = (int*)(bmax + 256);              // LOGITS_GRID
    int*   yid    = bidx + 256;                      // 1

    // encoder
    embed_kernel<<<T, 128, 0, stream>>>(x, enc_emb, xemb);
    gi0_wmma_kernel<<<dim3(T / 16, G3 / 64), 128, 0, stream>>>(xemb, enc_Wih0,
                                                               enc_bih0, gi0);
    enc_rnn_kernel<<<1, G3, ENC_SMEM_BYTES, stream>>>(gi0, enc_Whh0, enc_bhh0,
                                                      enc_Wih1, enc_bih1,
                                                      enc_Whh1, enc_bhh1, hstate);
    // decoder
    init_yid_kernel<<<1, 32, 0, stream>>>(sid, yid);
    for (int step = 0; step < STEPS; ++step) {
        dec_rnn_kernel<<<1, G3, 0, stream>>>(dec_emb, yid,
                                             dec_Wih0, dec_Whh0, dec_bih0, dec_bhh0,
                                             dec_Wih1, dec_Whh1, dec_bih1, dec_bhh1,
                                             hstate);
        logits_kernel<<<LOGITS_GRID, LOGITS_BLOCK, 0, stream>>>(out_W, out_b,
                                                                hstate, bmax, bidx);
        argmax_reduce_kernel<<<1, 256, 0, stream>>>(bmax, bidx, LOGITS_GRID,
                                                    yid, out, step);
    }
}